// GroupMambaLayer_90469191123241
// MI455X (gfx1250) — compile-verified
//
#include <hip/hip_runtime.h>
#include <math.h>

typedef __attribute__((ext_vector_type(2))) float v2f;
typedef __attribute__((ext_vector_type(8))) float v8f;

#define Bq 8
#define Cq 256
#define Gq 4
#define Dq 64
#define Lq 4096
#define NCHUNK 32
#define CHUNK 128

__device__ __forceinline__ float siluf(float v) { return v / (1.f + __expf(-v)); }
__device__ __forceinline__ float softplusf(float x) {
  return (x > 20.f) ? x : __logf(1.f + __expf(x));
}
// scan-order index for pixel (h,w) in direction/group g
__device__ __forceinline__ int lscan(int g, int h, int w) {
  int a = (h << 6) + w;
  int b2 = (w << 6) + h;
  if (g == 0) return a;
  if (g == 1) return (Lq - 1) - a;
  if (g == 2) return b2;
  return (Lq - 1) - b2;
}

// ---------------- K0a: p[b,c] = max over HxW ----------------
__global__ __launch_bounds__(256) void k_maxpool(const float* __restrict__ x,
                                                 float* __restrict__ p) {
  __shared__ float sm[256];
  int bc = blockIdx.x;
  const float* xr = x + (size_t)bc * Lq;
  int t = threadIdx.x;
  float m = -INFINITY;
  for (int k = 0; k < 16; ++k) m = fmaxf(m, xr[t + 256 * k]);
  sm[t] = m;
  __syncthreads();
  for (int s = 128; s > 0; s >>= 1) {
    if (t < s) sm[t] = fmaxf(sm[t], sm[t + s]);
    __syncthreads();
  }
  if (t == 0) p[bc] = sm[0];
}

// ---------------- K0b: channel-attention MLP -> z ----------------
__global__ __launch_bounds__(256) void k_ca(const float* __restrict__ p,
                                            const float* __restrict__ w1,
                                            const float* __restrict__ w2,
                                            float* __restrict__ z) {
  __shared__ float sp[256];
  __shared__ float sh[16];
  int b = blockIdx.x, t = threadIdx.x;
  sp[t] = p[b * 256 + t];
  __syncthreads();
  if (t < 16) {
    float s = 0.f;
    for (int c = 0; c < 256; ++c) s += sp[c] * w1[t * 256 + c];
    sh[t] = fmaxf(s, 0.f);
  }
  __syncthreads();
  float s = 0.f;
#pragma unroll
  for (int m = 0; m < 16; ++m) s += sh[m] * w2[t * 16 + m];
  z[b * 256 + t] = 1.f / (1.f + __expf(-s));
}

// ---------------- K1: in_proj GEMM (64pix x 64d) x (64d x 64k), f32 WMMA ---
__global__ __launch_bounds__(256) void k_inproj(const float* __restrict__ x,
                                                const float* __restrict__ w,
                                                float* __restrict__ u) {
  __shared__ float xl[64][65];  // xl[pix][kd]
  __shared__ float wl[64][65];  // wl[kd][ncol] = w[ncol][kd]
  int bi = blockIdx.x;
  int gb = bi >> 6;
  int tile = bi & 63;
  int g = gb >> 3, b = gb & 7;
  int pixbase = tile * 64;
  int t = threadIdx.x;
  const float* xb = x + ((size_t)b * Cq + g * Dq) * Lq + pixbase;
  const float* wg = w + (size_t)g * Dq * Dq;
  for (int k = 0; k < 16; ++k) {
    int i = t + 256 * k;
    int kd = i >> 6, pix = i & 63;
    xl[pix][kd] = xb[(size_t)kd * Lq + pix];
    wl[i & 63][i >> 6] = wg[i];
  }
  __syncthreads();
  int lid = t & 31, wv = t >> 5;
  int m = lid & 15, ks = (lid >> 4) * 2;
  int nt = wv & 3;
  int mt0 = wv >> 2;       // 0..1
  int mt1 = mt0 + 2;       // 2..3
  v8f acc0 = {}; v8f acc1 = {};
  for (int kk = 0; kk < 16; ++kk) {
    int k0 = kk * 4 + ks;
    v2f bf; bf.x = wl[k0][nt * 16 + m]; bf.y = wl[k0 + 1][nt * 16 + m];
    v2f a0; a0.x = xl[mt0 * 16 + m][k0]; a0.y = xl[mt0 * 16 + m][k0 + 1];
    v2f a1; a1.x = xl[mt1 * 16 + m][k0]; a1.y = xl[mt1 * 16 + m][k0 + 1];
    acc0 = __builtin_amdgcn_wmma_f32_16x16x4_f32(false, a0, false, bf, (short)0, acc0, false, false);
    acc1 = __builtin_amdgcn_wmma_f32_16x16x4_f32(false, a1, false, bf, (short)0, acc1, false, false);
  }
  float* ub = u + ((size_t)gb * Lq + pixbase) * 64;
  int col = nt * 16 + m;
  int moff = (lid >> 4) * 8;
#pragma unroll
  for (int r = 0; r < 8; ++r) {
    ub[(size_t)(mt0 * 16 + r + moff) * 64 + col] = acc0[r];
    ub[(size_t)(mt1 * 16 + r + moff) * 64 + col] = acc1[r];
  }
}

// ---------------- K2: depthwise 3x3 + SiLU + x_dbl, writes in scan order ---
__global__ __launch_bounds__(256) void k_conv(const float* __restrict__ u,
                                              const float* __restrict__ cw,
                                              const float* __restrict__ cb,
                                              const float* __restrict__ xw,
                                              float* __restrict__ us,
                                              float* __restrict__ xdbl) {
  __shared__ float sl[64][65];  // post-silu row [w][c]
  int bi = blockIdx.x;
  int gb = bi >> 6;
  int h = bi & 63;
  int g = gb >> 3;
  int t = threadIdx.x;
  int c = t & 63, wq = t >> 6;
  const float* ug = u + (size_t)gb * Lq * 64;
  float wr[9];
#pragma unroll
  for (int j = 0; j < 9; ++j) wr[j] = cw[((size_t)g * 64 + c) * 9 + j];
  float bias = cb[g * 64 + c];
  for (int wi = 0; wi < 16; ++wi) {
    int w = wi * 4 + wq;
    float s = bias;
    for (int dy = -1; dy <= 1; ++dy) {
      int hh = h + dy;
      if (hh < 0 || hh > 63) continue;
      for (int dx = -1; dx <= 1; ++dx) {
        int ww = w + dx;
        if (ww < 0 || ww > 63) continue;
        s += ug[((size_t)hh * 64 + ww) * 64 + c] * wr[(dy + 1) * 3 + (dx + 1)];
      }
    }
    float v = siluf(s);
    sl[w][c] = v;
    us[((size_t)gb * Lq + lscan(g, h, w)) * 64 + c] = v;
  }
  __syncthreads();
  {
    int w = t & 63, k = t >> 6;  // k in 0..3
    const float* xk = xw + ((size_t)g * 6 + k) * 64;
    float s = 0.f;
    for (int cc = 0; cc < 64; ++cc) s += sl[w][cc] * xk[cc];
    xdbl[((size_t)gb * Lq + lscan(g, h, w)) * 6 + k] = s;
    if (t < 128) {
      int k2 = 4 + (t >> 6);
      const float* xk2 = xw + ((size_t)g * 6 + k2) * 64;
      float s2 = 0.f;
      for (int cc = 0; cc < 64; ++cc) s2 += sl[w][cc] * xk2[cc];
      xdbl[((size_t)gb * Lq + lscan(g, h, w)) * 6 + k2] = s2;
    }
  }
}

// ---------------- K3a: per-chunk scan summaries (Aprod, Bacc) ----------------
__global__ __launch_bounds__(64) void k_scan_partial(const float* __restrict__ us,
                                                     const float* __restrict__ xdbl,
                                                     const float* __restrict__ dtw,
                                                     const float* __restrict__ dtb,
                                                     const float* __restrict__ alog,
                                                     float* __restrict__ cs) {
  int bi = blockIdx.x;
  int gb = bi / NCHUNK, ch = bi % NCHUNK;
  int g = gb >> 3;
  int c = threadIdx.x;
  const float4 w4 = reinterpret_cast<const float4*>(dtw)[(size_t)g * 64 + c];
  float bias = dtb[g * 64 + c];
  float A = -__expf(alog[g * 64 + c]);
  float Ap = 1.f, Bp = 0.f;
  size_t base = (size_t)gb * Lq + (size_t)ch * CHUNK;
  for (int tt = 0; tt < CHUNK; ++tt) {
    if ((tt & 15) == 0)
      __builtin_prefetch(us + (base + tt + 16) * 64 + c, 0, 0);
    const float* xd = xdbl + (base + tt) * 6;
    float dr = xd[0] * w4.x + xd[1] * w4.y + xd[2] * w4.z + xd[3] * w4.w + bias;
    float dtv = softplusf(dr);
    float a = __expf(dtv * A);
    float uv = us[(base + tt) * 64 + c];
    float bb = dtv * xd[4] * uv;
    Ap *= a;
    Bp = a * Bp + bb;
  }
  size_t o = (((size_t)gb * 64 + c) * NCHUNK + ch) * 2;
  cs[o] = Ap;
  cs[o + 1] = Bp;
}

// ---------------- K3b: combine chunk summaries into chunk-start prefixes ---
__global__ __launch_bounds__(64) void k_scan_prefix(const float* __restrict__ cs,
                                                    float* __restrict__ pref) {
  int gb = blockIdx.x;
  int c = threadIdx.x;
  size_t base = ((size_t)gb * 64 + c) * NCHUNK;
  float h = 0.f;
  for (int ch = 0; ch < NCHUNK; ++ch) {
    pref[base + ch] = h;
    h = cs[(base + ch) * 2] * h + cs[(base + ch) * 2 + 1];
  }
}

// ---------------- K3c: apply scan, produce ys ----------------
__global__ __launch_bounds__(64) void k_scan_apply(const float* __restrict__ us,
                                                   const float* __restrict__ xdbl,
                                                   const float* __restrict__ dtw,
                                                   const float* __restrict__ dtb,
                                                   const float* __restrict__ alog,
                                                   const float* __restrict__ dsk,
                                                   const float* __restrict__ pref,
                                                   float* __restrict__ ys) {
  int bi = blockIdx.x;
  int gb = bi / NCHUNK, ch = bi % NCHUNK;
  int g = gb >> 3;
  int c = threadIdx.x;
  const float4 w4 = reinterpret_cast<const float4*>(dtw)[(size_t)g * 64 + c];
  float bias = dtb[g * 64 + c];
  float A = -__expf(alog[g * 64 + c]);
  float Dv = dsk[g * 64 + c];
  float h = pref[((size_t)gb * 64 + c) * NCHUNK + ch];
  size_t base = (size_t)gb * Lq + (size_t)ch * CHUNK;
  for (int tt = 0; tt < CHUNK; ++tt) {
    if ((tt & 15) == 0)
      __builtin_prefetch(us + (base + tt + 16) * 64 + c, 0, 0);
    const float* xd = xdbl + (base + tt) * 6;
    float dr = xd[0] * w4.x + xd[1] * w4.y + xd[2] * w4.z + xd[3] * w4.w + bias;
    float dtv = softplusf(dr);
    float a = __expf(dtv * A);
    float uv = us[(base + tt) * 64 + c];
    h = a * h + dtv * xd[4] * uv;
    ys[(base + tt) * 64 + c] = h * xd[5] + Dv * uv;
  }
}

// ---------------- K4: outnorm + gate + out_proj WMMA + z + final LN + transpose
__global__ __launch_bounds__(256) void k_out(const float* __restrict__ ys,
                                             const float* __restrict__ yin,
                                             const float* __restrict__ ong,
                                             const float* __restrict__ onb,
                                             const float* __restrict__ opw,
                                             const float* __restrict__ z,
                                             const float* __restrict__ ng,
                                             const float* __restrict__ nb,
                                             float* __restrict__ out) {
  __shared__ float ol[32][65];    // per-group activations (GEMM A)
  __shared__ float wl[64][65];    // wl[k][dd] = out_proj_w[dd][k]
  __shared__ float xm[32][258];   // combined 256-channel panel
  __shared__ float red[32][8];
  __shared__ float red2[32][8];
  __shared__ float mv[32][2];
  int bi = blockIdx.x;
  int b = bi >> 7, tile = bi & 127;
  int p0 = tile * 32;
  int h = p0 >> 6, w0 = p0 & 63;
  int t = threadIdx.x;
  int lid = t & 31, wv = t >> 5;
  for (int g = 0; g < 4; ++g) {
    int gb = g * 8 + b;
    // load ys (undo scan permutation)
    for (int k = 0; k < 8; ++k) {
      int i = t + 256 * k;
      int pix = i >> 6, c = i & 63;
      int ls = lscan(g, h, w0 + pix);
      ol[pix][c] = ys[((size_t)gb * Lq + ls) * 64 + c];
    }
    __syncthreads();
    // mean/var over di per pixel
    {
      int pix = t >> 3, j = t & 7;
      float s = 0.f, ss = 0.f;
      for (int c = j; c < 64; c += 8) { float v = ol[pix][c]; s += v; ss += v * v; }
      red[pix][j] = s; red2[pix][j] = ss;
    }
    __syncthreads();
    if (t < 32) {
      float s = 0.f, ss = 0.f;
      for (int j = 0; j < 8; ++j) { s += red[t][j]; ss += red2[t][j]; }
      float m = s * (1.f / 64.f);
      float v = ss * (1.f / 64.f) - m * m;
      mv[t][0] = m; mv[t][1] = rsqrtf(v + 1e-5f);
    }
    __syncthreads();
    // normalize + SiLU(yg) gate
    for (int k = 0; k < 8; ++k) {
      int i = t + 256 * k;
      int pix = i & 31, c = i >> 5;
      float v = (ol[pix][c] - mv[pix][0]) * mv[pix][1] * ong[g * 64 + c] + onb[g * 64 + c];
      float yv = yin[((size_t)b * Cq + g * 64 + c) * Lq + p0 + pix];
      ol[pix][c] = v * siluf(yv);
    }
    // stage transposed out_proj weights
    for (int k = 0; k < 16; ++k) {
      int i = t + 256 * k;
      wl[i & 63][i >> 6] = opw[(size_t)g * 4096 + i];
    }
    __syncthreads();
    // WMMA: (32 x 64) x (64 x 64) -> xm columns g*64..g*64+63
    {
      int m = lid & 15, ks = (lid >> 4) * 2;
      int mt = wv >> 2, nt = wv & 3;
      v8f acc = {};
      for (int kk = 0; kk < 16; ++kk) {
        int k0 = kk * 4 + ks;
        v2f a;  a.x  = ol[mt * 16 + m][k0];      a.y  = ol[mt * 16 + m][k0 + 1];
        v2f bf; bf.x = wl[k0][nt * 16 + m];      bf.y = wl[k0 + 1][nt * 16 + m];
        acc = __builtin_amdgcn_wmma_f32_16x16x4_f32(false, a, false, bf, (short)0, acc, false, false);
      }
      int col = g * 64 + nt * 16 + m;
      int moff = (lid >> 4) * 8;
#pragma unroll
      for (int r = 0; r < 8; ++r) xm[mt * 16 + r + moff][col] = acc[r];
    }
    __syncthreads();
  }
  // x z, LayerNorm over C=256, transposed store
  {
    int pix = t >> 3, j = t & 7;
    float s = 0.f, ss = 0.f;
    for (int c = j; c < 256; c += 8) {
      float v = xm[pix][c] * z[b * 256 + c];
      xm[pix][c] = v;
      s += v; ss += v * v;
    }
    red[pix][j] = s; red2[pix][j] = ss;
  }
  __syncthreads();
  if (t < 32) {
    float s = 0.f, ss = 0.f;
    for (int j = 0; j < 8; ++j) { s += red[t][j]; ss += red2[t][j]; }
    float m = s * (1.f / 256.f);
    float v = ss * (1.f / 256.f) - m * m;
    mv[t][0] = m; mv[t][1] = rsqrtf(v + 1e-5f);
  }
  __syncthreads();
  {
    int pix = t & 31, cq = t >> 5;
    for (int ci = 0; ci < 32; ++ci) {
      int c = cq * 32 + ci;
      float v = (xm[pix][c] - mv[pix][0]) * mv[pix][1] * ng[c] + nb[c];
      out[((size_t)b * Cq + c) * Lq + p0 + pix] = v;
    }
  }
}

extern "C" void kernel_launch(void* const* d_in, const int* in_sizes, int n_in,
                              void* d_out, int out_size, void* d_ws, size_t ws_size,
                              hipStream_t stream) {
  const float* x    = (const float*)d_in[0];
  const float* y    = (const float*)d_in[1];
  const float* caw1 = (const float*)d_in[2];
  const float* caw2 = (const float*)d_in[3];
  const float* ipw  = (const float*)d_in[4];
  const float* cw   = (const float*)d_in[5];
  const float* cb   = (const float*)d_in[6];
  const float* xpw  = (const float*)d_in[7];
  const float* dtw  = (const float*)d_in[8];
  const float* dtb  = (const float*)d_in[9];
  const float* alog = (const float*)d_in[10];
  const float* dsk  = (const float*)d_in[11];
  const float* ong  = (const float*)d_in[12];
  const float* onb  = (const float*)d_in[13];
  const float* opw  = (const float*)d_in[14];
  const float* ng   = (const float*)d_in[15];
  const float* nb   = (const float*)d_in[16];
  float* ws = (float*)d_ws;

  // workspace layout (floats); ys reuses the u buffer (u dead after k_conv)
  float* p    = ws;                                  // 2048
  float* z    = p + 2048;                            // 2048
  float* u    = z + 2048;                            // 32*L*64 = 8.39M
  float* us   = u + (size_t)32 * Lq * 64;            // 8.39M
  float* xdbl = us + (size_t)32 * Lq * 64;           // 32*L*6
  float* cs   = xdbl + (size_t)32 * Lq * 6;          // 2048*NCHUNK*2
  float* pref = cs + (size_t)2048 * NCHUNK * 2;      // 2048*NCHUNK
  float* ys   = u;
  float* out  = (float*)d_out;

  k_maxpool<<<Bq * Cq, 256, 0, stream>>>(x, p);
  k_ca<<<Bq, 256, 0, stream>>>(p, caw1, caw2, z);
  k_inproj<<<Gq * Bq * (Lq / 64), 256, 0, stream>>>(x, ipw, u);
  k_conv<<<Gq * Bq * 64, 256, 0, stream>>>(u, cw, cb, xpw, us, xdbl);
  k_scan_partial<<<Gq * Bq * NCHUNK, 64, 0, stream>>>(us, xdbl, dtw, dtb, alog, cs);
  k_scan_prefix<<<Gq * Bq, 64, 0, stream>>>(cs, pref);
  k_scan_apply<<<Gq * Bq * NCHUNK, 64, 0, stream>>>(us, xdbl, dtw, dtb, alog, dsk, pref, ys);
  k_out<<<Bq * (Lq / 32), 256, 0, stream>>>(ys, y, ong, onb, opw, z, ng, nb, out);
}